// CombinedLoss_76630806495904
// MI455X (gfx1250) — compile-verified
//
#include <hip/hip_runtime.h>
#include <stdint.h>

#define INF_VAL 100000000.0f
#define MAX_M 512

// ---------------------------------------------------------------------------
// One pyramid level: one thread per anchor.
// Annotations (M x 3 f32) are staged into LDS via gfx1250 async global->LDS
// DMA (ASYNCcnt tracked, s_wait_asynccnt), then a per-block precompute packs
// the inner-loop invariants into a 16B-aligned float4 table so the hot loop
// is one ds_load_b128 (uniform-address broadcast) + ~10 VALU ops.
// ---------------------------------------------------------------------------
__global__ __launch_bounds__(256) void assign_level_kernel(
    const float* __restrict__ ann_g,     // [M*3] (l, r, cls)
    const float* __restrict__ anchors,   // [n]
    float* __restrict__ out,             // [12*A] concatenated outputs
    int M, int n, int levelBase, int A,
    float stride, float inv_stride,
    float lo, float hi, float level_out)
{
    __shared__ __align__(16) float s_ann[MAX_M * 3];  // raw annotations
    __shared__ __align__(16) float s_pre[MAX_M * 4];  // l, r, sub_hi, area

    const int tid = threadIdx.x;

    // ---- Stage annotations into LDS with async global->LDS B128 DMA ----
    const int total = 3 * M;             // floats
    const int nchunks = total >> 2;      // 16B chunks (512*3/4 = 384)
    const uint32_t lds_base = (uint32_t)(uintptr_t)&s_ann[0]; // low 32b of
                                                              // generic shared
                                                              // ptr == LDS off
    for (int c = tid; c < nchunks; c += blockDim.x) {
        const float* gaddr = ann_g + 4 * c;
        uint32_t loff = lds_base + 16u * c;
        asm volatile("global_load_async_to_lds_b128 %0, %1, off"
                     :
                     : "v"(loff), "v"(gaddr)
                     : "memory");
    }
    // Tail (only if 3*M not divisible by 4; not hit for M=512)
    for (int t = (nchunks << 2) + tid; t < total; t += blockDim.x) {
        s_ann[t] = ann_g[t];
    }
    asm volatile("s_wait_asynccnt 0" ::: "memory");
    __syncthreads();

    // ---- Precompute per-annotation invariants into float4 table ----
    for (int j = tid; j < M; j += blockDim.x) {
        float l = s_ann[3 * j + 0];
        float r = s_ann[3 * j + 1];
        float c = s_ann[3 * j + 2];
        // radius = (c==0)*DOWNBEAT_RADIUS + (c==1)*BEAT_RADIUS
        float radius = ((c == 0.0f) ? 4.5f : 0.0f) + ((c == 1.0f) ? 2.5f : 0.0f);
        float sub_hi = fminf(r, l + radius * stride);  // radius*stride exact
        s_pre[4 * j + 0] = l;
        s_pre[4 * j + 1] = r;
        s_pre[4 * j + 2] = sub_hi;
        s_pre[4 * j + 3] = r - l;                      // area
    }
    __syncthreads();

    const int i = blockIdx.x * blockDim.x + tid;
    if (i >= n) return;

    const float a = anchors[i];

    // ---- argmin over annotations (first-min semantics like jnp.argmin) ----
    float best = INF_VAL;
    int bj = 0;
    const float4* __restrict__ P = (const float4*)s_pre;  // 16B-aligned
    for (int j = 0; j < M; ++j) {
        float4 p = P[j];                  // ds_load_b128, uniform addr
        float lstar = a - p.x;
        float rstar = p.y - a;
        float m = fmaxf(lstar, rstar);
        bool ok = (a >= p.x) && (a <= p.z) && (m >= lo) && (m <= hi);
        float val = ok ? p.w : INF_VAL;
        if (val < best) { best = val; bj = j; }
    }

    const bool pos = (best != INF_VAL);
    const float al = s_ann[3 * bj + 0];
    const float ar = s_ann[3 * bj + 1];
    const float ac = pos ? s_ann[3 * bj + 2] : 0.0f;
    const float ls = a - al;
    const float rs = ar - a;

    const int g = levelBase + i;
    // Output layout (floats): pos[A] | assigned[A,3] | norm_ann[A,3] |
    //                         ls[A] | rs[A] | ls/s[A] | rs/s[A] | levels[A]
    out[g]                 = pos ? 1.0f : 0.0f;
    out[A + 3 * g + 0]     = al;
    out[A + 3 * g + 1]     = ar;
    out[A + 3 * g + 2]     = ac;
    out[4 * A + 3 * g + 0] = al * inv_stride;   // /2^k exact
    out[4 * A + 3 * g + 1] = ar * inv_stride;
    out[4 * A + 3 * g + 2] = ac;
    out[7 * A + g]         = ls;
    out[8 * A + g]         = rs;
    out[9 * A + g]         = ls * inv_stride;
    out[10 * A + g]        = rs * inv_stride;
    out[11 * A + g]        = level_out;
}

// Host: reproduce RANGES * TARGET_RATE in double, cast to f32 (bit-exact
// with the Python reference).
static void level_params(int lvl, float* lo, float* hi) {
    const double clusters[5] = {0.3, 0.6, 1.2, 2.4, 4.8};
    double edges[4];
    for (int i = 0; i < 4; ++i)
        edges[i] = clusters[i] + (clusters[i + 1] - clusters[i]) / 2.0;
    double ranges[5][2] = {
        {-1.0,     edges[0]},
        {edges[0], edges[1]},
        {edges[1], edges[2]},
        {edges[2], edges[3]},
        {edges[3], 1000.0}
    };
    const double TR = 22050.0 / 256.0;   // TARGET_RATE, exact in double
    *lo = (float)(ranges[lvl][0] * TR);
    *hi = (float)(ranges[lvl][1] * TR);
}

extern "C" void kernel_launch(void* const* d_in, const int* in_sizes, int n_in,
                              void* d_out, int out_size, void* d_ws, size_t ws_size,
                              hipStream_t stream) {
    (void)d_ws; (void)ws_size; (void)out_size;

    const float* ann = (const float*)d_in[0];
    float* out = (float*)d_out;

    const int M = in_sizes[0] / 3;       // 512
    int A = 0;
    for (int lvl = 0; lvl < 5 && (1 + lvl) < n_in; ++lvl) A += in_sizes[1 + lvl];

    int base = 0;
    for (int lvl = 0; lvl < 5; ++lvl) {
        const int n = in_sizes[1 + lvl];
        const float stride = (float)(1 << lvl);
        float lo, hi;
        level_params(lvl, &lo, &hi);
        const int threads = 256;
        const int blocks = (n + threads - 1) / threads;
        assign_level_kernel<<<blocks, threads, 0, stream>>>(
            ann, (const float*)d_in[1 + lvl], out,
            M, n, base, A,
            stride, 1.0f / stride,
            lo, hi, (float)(lvl + 1));
        base += n;
    }
}